// GATLinkAutoencoder_47433618817227
// MI455X (gfx1250) — compile-verified
//
#include <hip/hip_runtime.h>

// ---------------------------------------------------------------------------
// GAT link autoencoder for MI455X (gfx1250).
// Dense projections use V_WMMA_F32_16X16X4_F32 (exact f32 matrix path).
// Edge softmax/aggregation uses atomics (memory-bound part of the workload).
// ---------------------------------------------------------------------------

typedef float v2f __attribute__((ext_vector_type(2)));
typedef float v8f __attribute__((ext_vector_type(8)));

static inline int cdiv(int a, int b) { return (a + b - 1) / b; }

// ---------------- float <-> order-preserving uint (for atomic max) ----------
__device__ __forceinline__ unsigned f2ord(float f) {
    unsigned u = __float_as_uint(f);
    return (u & 0x80000000u) ? ~u : (u | 0x80000000u);
}
__device__ __forceinline__ float ord2f(unsigned u) {
    return (u & 0x80000000u) ? __uint_as_float(u & 0x7FFFFFFFu)
                             : __uint_as_float(~u);
}

// ---------------- WMMA GEMM: Out[M,Nout] = X[M,K] @ W[K,Nout] (+bias+resid) -
// One wave computes one 16x16 tile via V_WMMA_F32_16X16X4_F32, K in steps of 4.
// A 16x4 layout: lanes 0-15 row M=lane  (VGPR0=K0, VGPR1=K1),
//                lanes 16-31 row M=lane-16 (VGPR0=K2, VGPR1=K3).
// B 4x16 layout: lanes 0-15 col N=lane (v.x=K0,v.y=K1); lanes 16-31 (K2,K3).
// C/D: VGPR r -> row r (lanes 0-15) / row r+8 (lanes 16-31), col = lane&15.
__global__ void wmma_gemm_f32(const float* __restrict__ X,
                              const float* __restrict__ W,
                              float* __restrict__ Out,
                              const float* __restrict__ bias,
                              const float* __restrict__ resid,
                              int K, int Nout, int tiles_n, int total_tiles)
{
    const int wave = threadIdx.x >> 5;
    const int lane = threadIdx.x & 31;
    const int t    = blockIdx.x * (blockDim.x >> 5) + wave;
    if (t >= total_tiles) return;          // wave-uniform: EXEC stays all-1s

    const int tm   = t / tiles_n;
    const int tn   = t - tm * tiles_n;
    const int half = lane >> 4;            // 0: K0/K1, 1: K2/K3
    const int l15  = lane & 15;
    const int arow = tm * 16 + l15;
    const int bcol = tn * 16 + l15;

    v8f acc = {0.f, 0.f, 0.f, 0.f, 0.f, 0.f, 0.f, 0.f};

    for (int k = 0; k < K; k += 4) {
        const int ka = k + half * 2;
        v2f a = *(const v2f*)(X + (size_t)arow * K + ka);   // 8B aligned (K%4==0, ka even)
        v2f b;
        b.x = W[(size_t)ka * Nout + bcol];
        b.y = W[(size_t)(ka + 1) * Nout + bcol];
        acc = __builtin_amdgcn_wmma_f32_16x16x4_f32(false, a, false, b,
                                                    (short)0, acc, false, false);
    }

    const int rbase = tm * 16 + half * 8;
    const int cc    = tn * 16 + l15;
#pragma unroll
    for (int r = 0; r < 8; ++r) {
        const int rr = rbase + r;
        float v = acc[r];
        if (bias)  v += bias[cc];
        if (resid) v += resid[(size_t)rr * Nout + cc];
        Out[(size_t)rr * Nout + cc] = v;
    }
}

// ---------------- per-(node,head) attention logits --------------------------
__global__ void alpha_kernel(const float* __restrict__ h,
                             const float* __restrict__ a_s,
                             const float* __restrict__ a_d,
                             float* __restrict__ asrc, float* __restrict__ adst,
                             int total, int H, int C)
{
    int idx = blockIdx.x * blockDim.x + threadIdx.x;   // idx = node*H + head
    if (idx >= total) return;
    const int hh = idx % H;
    const float* hp = h + (size_t)idx * C;
    const float* as = a_s + hh * C;
    const float* ad = a_d + hh * C;
    float s = 0.f, d = 0.f;
    for (int c = 0; c < C; ++c) {
        float v = hp[c];
        s = fmaf(v, as[c], s);
        d = fmaf(v, ad[c], d);
    }
    asrc[idx] = s;
    adst[idx] = d;
}

// ---------------- edge pass 1: e = lrelu0.2(as[src]+ad[dst]); segment max ---
__global__ void edge_max_kernel(const int* __restrict__ ei, int E, int Etot, int H,
                                const float* __restrict__ asrc,
                                const float* __restrict__ adst,
                                float* __restrict__ ebuf,
                                unsigned* __restrict__ menc)
{
    int idx = blockIdx.x * blockDim.x + threadIdx.x;   // idx = edge*H + head
    if (idx >= Etot * H) return;
    const int e = idx / H;
    const int h = idx - e * H;
    int s, d;
    if (e < E) { s = ei[e]; d = ei[E + e]; } else { s = d = e - E; }
    float v = asrc[s * H + h] + adst[d * H + h];
    v = v > 0.f ? v : 0.2f * v;
    ebuf[idx] = v;
    atomicMax(&menc[d * H + h], f2ord(v));
}

// ---------------- decode encoded max in place (0 marker -> 0.0, per ref) ----
__global__ void decode_max_kernel(unsigned* __restrict__ m, int total)
{
    int idx = blockIdx.x * blockDim.x + threadIdx.x;
    if (idx >= total) return;
    unsigned u = m[idx];
    float v = (u == 0u) ? 0.f : ord2f(u);
    ((float*)m)[idx] = v;
}

// ---------------- edge pass 2: ex = exp(e - m[dst]); segment sum ------------
__global__ void edge_exp_kernel(const int* __restrict__ ei, int E, int Etot, int H,
                                float* __restrict__ ebuf,
                                const float* __restrict__ m,
                                float* __restrict__ denom)
{
    int idx = blockIdx.x * blockDim.x + threadIdx.x;
    if (idx >= Etot * H) return;
    const int e = idx / H;
    const int h = idx - e * H;
    const int d = (e < E) ? ei[E + e] : (e - E);
    float ex = expf(ebuf[idx] - m[d * H + h]);
    ebuf[idx] = ex;
    atomicAdd(&denom[d * H + h], ex);
}

// ---------------- edge pass 3: out[dst] += coef * h[src] --------------------
__global__ void edge_aggr_kernel(const int* __restrict__ ei, int E, int H, int C,
                                 const float* __restrict__ ebuf,
                                 const float* __restrict__ denom,
                                 const float* __restrict__ hfeat,
                                 float* __restrict__ out, int total)
{
    int idx = blockIdx.x * blockDim.x + threadIdx.x;   // idx = edge*(H*C)+feat
    if (idx >= total) return;
    const int HC = H * C;
    const int e  = idx / HC;
    const int f  = idx - e * HC;
    const int h  = f / C;
    int s, d;
    if (e < E) { s = ei[e]; d = ei[E + e]; } else { s = d = e - E; }
    const float coef = ebuf[e * H + h] / (denom[d * H + h] + 1e-16f);
    atomicAdd(&out[(size_t)d * HC + f], coef * hfeat[(size_t)s * HC + f]);
}

// ---------------- bias + optional LeakyReLU(0.01) ---------------------------
__global__ void finalize_kernel(float* __restrict__ out,
                                const float* __restrict__ b,
                                int total, int F, int act)
{
    int idx = blockIdx.x * blockDim.x + threadIdx.x;
    if (idx >= total) return;
    const int f = idx % F;
    float v = out[idx] + b[f];
    if (act) v = v > 0.f ? v : 0.01f * v;
    out[idx] = v;
}

// ---------------- one full GAT layer ----------------------------------------
static void run_gat_layer(const float* xin, int fin,
                          const float* W, const float* a_s, const float* a_d,
                          const float* b, int H, int C,
                          const int* ei, int E, int Etot, int n,
                          float* hbuf, float* ebuf, float* mbuf, float* denom,
                          float* asrc, float* adst, float* out, bool act,
                          hipStream_t stream)
{
    const int HC = H * C;
    const int tiles_n = HC / 16;
    const int total_tiles = (n / 16) * tiles_n;

    // GEMM first so `out` may alias `xin` (stream-ordered).
    wmma_gemm_f32<<<cdiv(total_tiles, 8), 256, 0, stream>>>(
        xin, W, hbuf, nullptr, nullptr, fin, HC, tiles_n, total_tiles);

    hipMemsetAsync(out,   0, (size_t)n * HC * sizeof(float), stream);
    hipMemsetAsync(mbuf,  0, (size_t)n * H  * sizeof(float), stream);
    hipMemsetAsync(denom, 0, (size_t)n * H  * sizeof(float), stream);

    alpha_kernel<<<cdiv(n * H, 256), 256, 0, stream>>>(
        hbuf, a_s, a_d, asrc, adst, n * H, H, C);

    edge_max_kernel<<<cdiv(Etot * H, 256), 256, 0, stream>>>(
        ei, E, Etot, H, asrc, adst, ebuf, (unsigned*)mbuf);

    decode_max_kernel<<<cdiv(n * H, 256), 256, 0, stream>>>(
        (unsigned*)mbuf, n * H);

    edge_exp_kernel<<<cdiv(Etot * H, 256), 256, 0, stream>>>(
        ei, E, Etot, H, ebuf, mbuf, denom);

    const int aggr_total = Etot * HC;   // max 850k*128 ≈ 1.09e8 < 2^31
    edge_aggr_kernel<<<cdiv(aggr_total, 256), 256, 0, stream>>>(
        ei, E, H, C, ebuf, denom, hbuf, out, aggr_total);

    finalize_kernel<<<cdiv(n * HC, 256), 256, 0, stream>>>(
        out, b, n * HC, HC, act ? 1 : 0);
}

// ---------------------------------------------------------------------------
extern "C" void kernel_launch(void* const* d_in, const int* in_sizes, int n_in,
                              void* d_out, int out_size, void* d_ws, size_t ws_size,
                              hipStream_t stream)
{
    const int IN_DIM = 128;

    const float* x    = (const float*)d_in[0];
    const int*   ei   = (const int*)d_in[1];
    const float* W1   = (const float*)d_in[2];
    const float* a1s  = (const float*)d_in[3];
    const float* a1d  = (const float*)d_in[4];
    const float* b1   = (const float*)d_in[5];
    const float* W2   = (const float*)d_in[6];
    const float* a2s  = (const float*)d_in[7];
    const float* a2d  = (const float*)d_in[8];
    const float* b2   = (const float*)d_in[9];
    const float* Wl   = (const float*)d_in[10];
    const float* als  = (const float*)d_in[11];
    const float* ald  = (const float*)d_in[12];
    const float* bl   = (const float*)d_in[13];
    const float* Wd   = (const float*)d_in[14];
    const float* ads_ = (const float*)d_in[15];
    const float* add_ = (const float*)d_in[16];
    const float* bd   = (const float*)d_in[17];
    const float* Wdec = (const float*)d_in[18];
    const float* bdec = (const float*)d_in[19];

    const int n    = in_sizes[0] / IN_DIM;   // 50000 (divisible by 16)
    const int E    = in_sizes[1] / 2;        // 800000
    const int Etot = E + n;                  // self-loops appended

    // ---- workspace carve-out (~75 MB) ----
    char* ws = (char*)d_ws;
    float* bufA  = (float*)ws; ws += (size_t)n * 128 * sizeof(float);
    float* hbuf  = (float*)ws; ws += (size_t)n * 128 * sizeof(float);
    float* ebuf  = (float*)ws; ws += (size_t)Etot * 4 * sizeof(float);
    float* asrc  = (float*)ws; ws += (size_t)n * 4 * sizeof(float);
    float* adst  = (float*)ws; ws += (size_t)n * 4 * sizeof(float);
    float* mbuf  = (float*)ws; ws += (size_t)n * 4 * sizeof(float);
    float* denom = (float*)ws; ws += (size_t)n * 4 * sizeof(float);
    float* xr    = (float*)ws; ws += (size_t)n * 32 * sizeof(float);

    float* recon = (float*)d_out;            // [n,128]
    float* z     = recon + (size_t)n * 128;  // [n,32]  (second ref output)

    // Layer 1: [n,128] -> [n,4,32] concat -> [n,128], lrelu 0.01
    run_gat_layer(x, 128, W1, a1s, a1d, b1, 4, 32, ei, E, Etot, n,
                  hbuf, ebuf, mbuf, denom, asrc, adst, bufA, true, stream);
    // Layer 2: [n,128] -> [n,128], lrelu 0.01  (in-place buffer reuse is safe)
    run_gat_layer(bufA, 128, W2, a2s, a2d, b2, 4, 32, ei, E, Etot, n,
                  hbuf, ebuf, mbuf, denom, asrc, adst, bufA, true, stream);
    // Latent: [n,128] -> z [n,32], no activation; write straight into d_out
    run_gat_layer(bufA, 128, Wl, als, ald, bl, 1, 32, ei, E, Etot, n,
                  hbuf, ebuf, mbuf, denom, asrc, adst, z, false, stream);
    // Decode GAT: z [n,32] -> xr [n,32], lrelu 0.01
    run_gat_layer(z, 32, Wd, ads_, add_, bd, 1, 32, ei, E, Etot, n,
                  hbuf, ebuf, mbuf, denom, asrc, adst, xr, true, stream);

    // Decoder GEMM with fused epilogue: recon = xr @ Wdec + bdec + x
    {
        const int tiles_n = 128 / 16;
        const int total_tiles = (n / 16) * tiles_n;
        wmma_gemm_f32<<<cdiv(total_tiles, 8), 256, 0, stream>>>(
            xr, Wdec, recon, bdec, x, 32, 128, tiles_n, total_tiles);
    }
}